// Mamba_3453153706765
// MI455X (gfx1250) — compile-verified
//
#include <hip/hip_runtime.h>
#include <hip/hip_bf16.h>
#include <cstdint>

// ---- problem constants (from reference) ----
#define B_SZ   2
#define LSEQ   2048
#define DMODEL 1024
#define DINNER 2048
#define DSTATE 16
#define DTRANK 64
#define XDBL_W 96              // DT_RANK + 2*D_STATE
#define NCHUNK 32
#define CLEN   (LSEQ / NCHUNK) // 64

typedef float v2f __attribute__((ext_vector_type(2)));
typedef float v8f __attribute__((ext_vector_type(8)));
typedef unsigned int v4u __attribute__((ext_vector_type(4)));
typedef int v4i __attribute__((ext_vector_type(4)));
typedef int v8i __attribute__((ext_vector_type(8)));

#if defined(__has_builtin)
#if __has_builtin(__builtin_amdgcn_tensor_load_to_lds) && __has_builtin(__builtin_amdgcn_s_wait_tensorcnt)
#define HAVE_TDM 1
#endif
#endif
#ifndef HAVE_TDM
#define HAVE_TDM 0
#endif

__device__ __forceinline__ float softplus_f(float x) {
  return (x > 20.0f) ? x : log1pf(__expf(x));
}
__device__ __forceinline__ float silu_f(float x) {
  return x / (1.0f + __expf(-x));
}

#if HAVE_TDM
// ---------------------------------------------------------------------------
// TDM: DMA a 2D tile (tile_dim1 rows x tile_dim0 elems, fp32, row stride
// `stride` elems) from global memory into contiguous LDS.  Issued by one wave;
// caller must __syncthreads() afterwards (the issuing wave waits TENSORcnt==0
// before reaching the barrier, so the tile is visible to the whole block).
// D# layout per CDNA5 ISA ch.8 (async_tensor).
// ---------------------------------------------------------------------------
__device__ __forceinline__ void tdm_load_tile_2d(const float* gsrc, void* lds_dst,
                                                 unsigned width, unsigned rows,
                                                 unsigned stride) {
  const unsigned long long ga = (unsigned long long)(uintptr_t)gsrc;
  const unsigned lds_off = (unsigned)(uintptr_t)lds_dst;  // LDS aperture: addr[31:0]
  v4u g0;
  g0.x = 1u;                                             // count=1, user D#
  g0.y = lds_off;                                        // lds_addr [63:32]
  g0.z = (unsigned)(ga & 0xFFFFFFFFu);                   // global_addr[31:0]
  g0.w = (unsigned)((ga >> 32) & 0x01FFFFFFu) | (2u << 30); // global_addr[56:32], type=2
  v8i g1;
  g1[0] = (int)(2u << 16);            // workgroup_mask=0, data_size=2 (4 bytes)
  g1[1] = (int)(width << 16);         // tensor_dim0[15:0]  @ bits 63:48
  g1[2] = (int)(rows << 16);          // tensor_dim1[15:0]  @ bits 95:80
  g1[3] = (int)(width << 16);         // tile_dim0          @ bits 127:112
  g1[4] = (int)rows;                  // tile_dim1 (tile_dim2=0)
  g1[5] = (int)stride;                // tensor_dim0_stride[31:0]
  g1[6] = 0;                          // stride[47:32]=0, tensor_dim1_stride lo=0
  g1[7] = 0;
  v4i z4 = {0, 0, 0, 0};
#if __clang_major__ >= 23
  v8i z8 = {0, 0, 0, 0, 0, 0, 0, 0};
  __builtin_amdgcn_tensor_load_to_lds(g0, g1, z4, z4, z8, 0);
#else
  __builtin_amdgcn_tensor_load_to_lds(g0, g1, z4, z4, 0);
#endif
  __builtin_amdgcn_s_wait_tensorcnt(0);
}
#endif

// Stage B/C rows of XDBL for one (batch, chunk) into LDS.
__device__ __forceinline__ void stage_bc(float* sBC, const float* __restrict__ XDBL,
                                         int b, int l0) {
  const float* src = XDBL + (size_t)(b * LSEQ + l0) * XDBL_W + DTRANK;
#if HAVE_TDM
  if (threadIdx.x < 32) {                    // one wave issues the DMA
    tdm_load_tile_2d(src, sBC, 32u, (unsigned)CLEN, (unsigned)XDBL_W);
  }
#else
  for (int i = threadIdx.x; i < CLEN * 32; i += 256) {
    const int li = i >> 5, j = i & 31;       // j<16: B, j>=16: C
    sBC[i] = src[(size_t)li * XDBL_W + j];
  }
#endif
  __syncthreads();
}

// ---------------------------------------------------------------------------
// Generic fp32 WMMA GEMM: C[M,N] = A[M,K(lda)] * B[K,N(ldb)] (+bias)
// One wave computes a 16 x (NT*16) tile; WMMA shape 16x16x4 fp32.
// A frag (16x4): lanes 0-15 -> (K=k0,k0+1), lanes 16-31 -> (K=k0+2,k0+3), M=lane&15
// B frag (4x16): mirror layout, N=lane&15
// C/D: VGPR v -> M = v + 8*(lane>=16), N = lane&15
// ---------------------------------------------------------------------------
template<int NT, int WAVES, bool HAS_BIAS>
__global__ void __launch_bounds__(WAVES * 32)
gemm_f32_wmma(const float* __restrict__ A, const float* __restrict__ Bm,
              const float* __restrict__ bias, float* __restrict__ C,
              int K, int lda, int ldb, int ldc) {
  const int lane = threadIdx.x & 31;
  const int wid  = threadIdx.x >> 5;
  const int m15  = lane & 15;
  const int hi   = lane >> 4;        // 0 or 1
  const int kph  = hi << 1;          // K phase within group of 4
  const int row  = blockIdx.y * 16 + m15;
  const int nbase = (blockIdx.x * WAVES + wid) * (NT * 16);

  v8f acc[NT] = {};
  const float* Arow = A + (size_t)row * lda;
  for (int k0 = 0; k0 < K; k0 += 4) {
    v2f a;
    a.x = Arow[k0 + kph];
    a.y = Arow[k0 + kph + 1];
    const float* Brow = Bm + (size_t)(k0 + kph) * ldb;
#pragma unroll
    for (int t = 0; t < NT; ++t) {
      const int nc = nbase + t * 16 + m15;
      v2f b;
      b.x = Brow[nc];
      b.y = Brow[ldb + nc];
      acc[t] = __builtin_amdgcn_wmma_f32_16x16x4_f32(
          false, a, false, b, (short)0, acc[t], false, false);
    }
  }
  const int mbase = blockIdx.y * 16 + hi * 8;
#pragma unroll
  for (int t = 0; t < NT; ++t) {
    const int nc = nbase + t * 16 + m15;
    const float bv = HAS_BIAS ? bias[nc] : 0.0f;
#pragma unroll
    for (int v = 0; v < 8; ++v) {
      C[(size_t)(mbase + v) * ldc + nc] = acc[t][v] + bv;
    }
  }
}

// ---------------------------------------------------------------------------
// GEMM for the input projection: T = H @ W_in, split even/odd columns into
// separate X / Z buffers (xz[...,0::2] -> x, xz[...,1::2] -> z).
// ---------------------------------------------------------------------------
template<int NT, int WAVES>
__global__ void __launch_bounds__(WAVES * 32)
gemm_xz_wmma(const float* __restrict__ A, const float* __restrict__ Bm,
             float* __restrict__ Xb, float* __restrict__ Zb,
             int K, int lda, int ldb) {
  const int lane = threadIdx.x & 31;
  const int wid  = threadIdx.x >> 5;
  const int m15  = lane & 15;
  const int hi   = lane >> 4;
  const int kph  = hi << 1;
  const int row  = blockIdx.y * 16 + m15;
  const int nbase = (blockIdx.x * WAVES + wid) * (NT * 16);

  v8f acc[NT] = {};
  const float* Arow = A + (size_t)row * lda;
  for (int k0 = 0; k0 < K; k0 += 4) {
    v2f a;
    a.x = Arow[k0 + kph];
    a.y = Arow[k0 + kph + 1];
    const float* Brow = Bm + (size_t)(k0 + kph) * ldb;
#pragma unroll
    for (int t = 0; t < NT; ++t) {
      const int nc = nbase + t * 16 + m15;
      v2f b;
      b.x = Brow[nc];
      b.y = Brow[ldb + nc];
      acc[t] = __builtin_amdgcn_wmma_f32_16x16x4_f32(
          false, a, false, b, (short)0, acc[t], false, false);
    }
  }
  const int mbase = blockIdx.y * 16 + hi * 8;
#pragma unroll
  for (int t = 0; t < NT; ++t) {
    const int cg = nbase + t * 16 + m15;       // global column in [0, 2*DINNER)
    float* dst = (cg & 1) ? Zb : Xb;
    const int cc = cg >> 1;
#pragma unroll
    for (int v = 0; v < 8; ++v) {
      dst[(size_t)(mbase + v) * DINNER + cc] = acc[t][v];
    }
  }
}

// ---------------------------------------------------------------------------
// Depthwise conv1d along L (k=4, SAME pad (1,2)) + bias + SiLU
// ---------------------------------------------------------------------------
__global__ void __launch_bounds__(256)
conv_silu_kernel(const float* __restrict__ X, const float* __restrict__ Wc,
                 const float* __restrict__ bc, float* __restrict__ XC) {
  const int idx = blockIdx.x * 256 + threadIdx.x;   // over B*L*DINNER
  const int d  = idx & (DINNER - 1);
  const int l  = (idx >> 11) & (LSEQ - 1);
  const int bl = idx >> 11;                         // b*L + l
  float acc = bc[d];
#pragma unroll
  for (int w = 0; w < 4; ++w) {
    const int ls = l + w - 1;
    if (ls >= 0 && ls < LSEQ) {
      acc += X[(size_t)(bl + w - 1) * DINNER + d] * Wc[w * DINNER + d];
    }
  }
  XC[(size_t)idx] = silu_f(acc);
}

// ---------------------------------------------------------------------------
// Chunked associative scan.  Operator (a,b): s -> a*s + b composes as
// (a2,b2) o (a1,b1) = (a2*a1, a2*b1 + b2).
// Pass 1: per chunk, P = prod(dA), S = local state with zero init.
// Pass 2: serial prefix over the 32 chunks -> incoming state per chunk.
// Pass 3: re-run chunk seeded with incoming state, emit y.
// Block = 256 threads over d; all threads share one (b, chunk) so B/C rows
// are staged into LDS once per block via the Tensor Data Mover.
// ---------------------------------------------------------------------------
__global__ void __launch_bounds__(256)
scan_chunk_pass1(const float* __restrict__ DT, const float* __restrict__ XC,
                 const float* __restrict__ XDBL, const float* __restrict__ Alog,
                 float* __restrict__ P, float* __restrict__ S) {
  __shared__ float sBC[CLEN * 32];
  const int tid  = threadIdx.x;
  const int dblk = blockIdx.x & 7;                 // DINNER/256 = 8
  const int c    = (blockIdx.x >> 3) & (NCHUNK - 1);
  const int b    = blockIdx.x >> 8;
  const int d    = dblk * 256 + tid;
  const int l0   = c * CLEN;
  stage_bc(sBC, XDBL, b, l0);

  float An[DSTATE], Pr[DSTATE], St[DSTATE];
#pragma unroll
  for (int n = 0; n < DSTATE; ++n) {
    An[n] = -__expf(Alog[d * DSTATE + n]);
    Pr[n] = 1.0f;
    St[n] = 0.0f;
  }
  for (int i = 0; i < CLEN; ++i) {
    const size_t ridx = (size_t)(b * LSEQ + l0 + i);
    const float delta = softplus_f(DT[ridx * DINNER + d]);
    const float dx    = delta * XC[ridx * DINNER + d];
    const float* bc   = &sBC[i * 32];
#pragma unroll
    for (int n = 0; n < DSTATE; ++n) {
      const float dA = __expf(delta * An[n]);
      Pr[n] *= dA;
      St[n] = dA * St[n] + bc[n] * dx;
    }
  }
  const size_t base = ((size_t)(b * NCHUNK + c) * DINNER + d) * DSTATE;
#pragma unroll
  for (int n = 0; n < DSTATE; ++n) {
    P[base + n] = Pr[n];
    S[base + n] = St[n];
  }
}

__global__ void __launch_bounds__(256)
scan_chunk_pass2(const float* __restrict__ P, const float* __restrict__ S,
                 float* __restrict__ SIN) {
  const int t = blockIdx.x * 256 + threadIdx.x;    // over B*DINNER*DSTATE
  const int n = t & 15;
  const int d = (t >> 4) & (DINNER - 1);
  const int b = t >> 15;
  float s = 0.0f;
  for (int c = 0; c < NCHUNK; ++c) {
    const size_t base = ((size_t)(b * NCHUNK + c) * DINNER + d) * DSTATE + n;
    SIN[base] = s;
    s = P[base] * s + S[base];
  }
}

__global__ void __launch_bounds__(256)
scan_chunk_pass3(const float* __restrict__ DT, const float* __restrict__ XC,
                 const float* __restrict__ XDBL, const float* __restrict__ Alog,
                 const float* __restrict__ SIN, float* __restrict__ Y) {
  __shared__ float sBC[CLEN * 32];
  const int tid  = threadIdx.x;
  const int dblk = blockIdx.x & 7;
  const int c    = (blockIdx.x >> 3) & (NCHUNK - 1);
  const int b    = blockIdx.x >> 8;
  const int d    = dblk * 256 + tid;
  const int l0   = c * CLEN;
  stage_bc(sBC, XDBL, b, l0);

  const size_t base = ((size_t)(b * NCHUNK + c) * DINNER + d) * DSTATE;
  float An[DSTATE], St[DSTATE];
#pragma unroll
  for (int n = 0; n < DSTATE; ++n) {
    An[n] = -__expf(Alog[d * DSTATE + n]);
    St[n] = SIN[base + n];
  }
  for (int i = 0; i < CLEN; ++i) {
    const size_t ridx = (size_t)(b * LSEQ + l0 + i);
    const float delta = softplus_f(DT[ridx * DINNER + d]);
    const float dx    = delta * XC[ridx * DINNER + d];
    const float* bc   = &sBC[i * 32];
    float y = 0.0f;
#pragma unroll
    for (int n = 0; n < DSTATE; ++n) {
      const float dA = __expf(delta * An[n]);
      St[n] = dA * St[n] + bc[n] * dx;
      y += St[n] * bc[16 + n];
    }
    Y[ridx * DINNER + d] = y;
  }
}

// ---------------------------------------------------------------------------
// y = (y + x*D) * silu(z), in place on Y
// ---------------------------------------------------------------------------
__global__ void __launch_bounds__(256)
gate_kernel(const float* __restrict__ XC, const float* __restrict__ Z,
            const float* __restrict__ Dp, float* __restrict__ Y) {
  const size_t i = (size_t)blockIdx.x * 256 + threadIdx.x;
  const int d = (int)(i & (DINNER - 1));
  const float y = Y[i] + XC[i] * Dp[d];
  Y[i] = y * silu_f(Z[i]);
}

// ---------------------------------------------------------------------------
extern "C" void kernel_launch(void* const* d_in, const int* in_sizes, int n_in,
                              void* d_out, int out_size, void* d_ws, size_t ws_size,
                              hipStream_t stream) {
  (void)in_sizes; (void)n_in; (void)out_size; (void)ws_size;
  const float* H    = (const float*)d_in[0];
  const float* Win  = (const float*)d_in[1];
  const float* Wc   = (const float*)d_in[2];
  const float* bc   = (const float*)d_in[3];
  const float* Wx   = (const float*)d_in[4];
  const float* Wdt  = (const float*)d_in[5];
  const float* bdt  = (const float*)d_in[6];
  const float* Alog = (const float*)d_in[7];
  const float* Dp   = (const float*)d_in[8];
  const float* Wout = (const float*)d_in[9];
  float* Out = (float*)d_out;

  char* ws = (char*)d_ws;
  const size_t SZ_ACT  = (size_t)B_SZ * LSEQ * DINNER * sizeof(float);            // 32 MB
  const size_t SZ_XDBL = (size_t)B_SZ * LSEQ * XDBL_W * sizeof(float);            // 1.5 MB
  const size_t SZ_PS   = (size_t)B_SZ * NCHUNK * DINNER * DSTATE * sizeof(float); // 8 MB
  float* X    = (float*)(ws);                         // pre-conv x; reused as DT later
  float* Z    = (float*)(ws + SZ_ACT);
  float* XC   = (float*)(ws + 2 * SZ_ACT);
  float* XDBL = (float*)(ws + 3 * SZ_ACT);
  float* Y    = (float*)(ws + 3 * SZ_ACT + SZ_XDBL);
  float* P    = (float*)(ws + 4 * SZ_ACT + SZ_XDBL);
  float* S    = (float*)(ws + 4 * SZ_ACT + SZ_XDBL + SZ_PS);
  float* SIN  = (float*)(ws + 4 * SZ_ACT + SZ_XDBL + 2 * SZ_PS);
  float* DTb  = X;   // dt buffer reuses X once the conv has consumed it

  const int MROWS = B_SZ * LSEQ;   // 4096

  // 1) xz = H @ W_in  (M=4096, K=1024, N=4096) -> split into X / Z
  {
    dim3 grid((2 * DINNER) / 256, MROWS / 16);
    gemm_xz_wmma<4, 4><<<grid, 128, 0, stream>>>(H, Win, X, Z,
                                                 DMODEL, DMODEL, 2 * DINNER);
  }
  // 2) depthwise conv + SiLU: X -> XC
  conv_silu_kernel<<<(MROWS * DINNER) / 256, 256, 0, stream>>>(X, Wc, bc, XC);
  // 3) XDBL = XC @ W_x  (K=2048, N=96)
  {
    dim3 grid(1, MROWS / 16);
    gemm_f32_wmma<3, 2, false><<<grid, 64, 0, stream>>>(XC, Wx, nullptr, XDBL,
                                                        DINNER, DINNER, XDBL_W, XDBL_W);
  }
  // 4) DT = XDBL[:, :64] @ W_dt + b_dt  (K=64, N=2048) -> into X (reused)
  {
    dim3 grid(DINNER / 256, MROWS / 16);
    gemm_f32_wmma<4, 4, true><<<grid, 128, 0, stream>>>(XDBL, Wdt, bdt, DTb,
                                                        DTRANK, XDBL_W, DINNER, DINNER);
  }
  // 5) chunked selective scan -> Y
  scan_chunk_pass1<<<B_SZ * NCHUNK * (DINNER / 256), 256, 0, stream>>>(DTb, XC, XDBL, Alog, P, S);
  scan_chunk_pass2<<<(B_SZ * DINNER * DSTATE) / 256, 256, 0, stream>>>(P, S, SIN);
  scan_chunk_pass3<<<B_SZ * NCHUNK * (DINNER / 256), 256, 0, stream>>>(DTb, XC, XDBL, Alog, SIN, Y);
  // 6) gate: Y = (Y + XC*D) * silu(Z)
  gate_kernel<<<(MROWS * DINNER) / 256, 256, 0, stream>>>(XC, Z, Dp, Y);
  // 7) Out = Y @ W_out  (K=2048, N=1024)
  {
    dim3 grid(DMODEL / 256, MROWS / 16);
    gemm_f32_wmma<4, 4, false><<<grid, 128, 0, stream>>>(Y, Wout, nullptr, Out,
                                                         DINNER, DINNER, DMODEL, DMODEL);
  }
}